// WhisperCrossAttentionStepTVM_76227079569644
// MI455X (gfx1250) — compile-verified
//
#include <hip/hip_runtime.h>
#include <hip/hip_bf16.h>
#include <math.h>

// ---------------------------------------------------------------------------
// Whisper cross-attention decode step, MI455X (gfx1250, wave32).
// HBM-bound (~97MB moved, ~50MFLOP): flash-decoding split for bandwidth,
// v_wmma_f32_16x16x4_f32 for the QK^T scores (compute is free, WMMA gives a
// free score broadcast across lane halves via row-replicated A).
// ---------------------------------------------------------------------------

#define EMBED_DIM 1280
#define NUM_HEADS 20
#define HEAD_DIM  64
#define ENC_LEN   8192
#define ATTN_SCALE 0.125f          // 64^-0.5

#define NCHUNK 16                  // key chunks per head
#define WAVES_PER_BLOCK 8
#define PART (NCHUNK * WAVES_PER_BLOCK)              // 128 partials per head
#define KEYS_PER_WAVE (ENC_LEN / PART)               // 64
#define TILES_PER_WAVE (KEYS_PER_WAVE / 16)          // 4

typedef __attribute__((ext_vector_type(2))) float v2f;
typedef __attribute__((ext_vector_type(8))) float v8f;

// ---------------------------------------------------------------------------
// GEMV: y[row] = (sum_j x[j]*W[row,j] + b[row]) * scale   (1280x1280)
// grid = 1280 blocks, block = 128 threads; float2 coalesced row loads.
// ---------------------------------------------------------------------------
__global__ void gemv1280_kernel(const float* __restrict__ x,
                                const float* __restrict__ W,
                                const float* __restrict__ b,
                                float* __restrict__ y,
                                float scale) {
    const int row = blockIdx.x;
    const int tid = threadIdx.x;
    const float2* __restrict__ wr = reinterpret_cast<const float2*>(W + (size_t)row * EMBED_DIM);
    const float2* __restrict__ x2 = reinterpret_cast<const float2*>(x);
    float sum = 0.f;
#pragma unroll
    for (int j = 0; j < 5; ++j) {
        const int idx = tid + j * 128;           // 640 float2 per row
        const float2 w2 = wr[idx];
        const float2 xv = x2[idx];
        sum = fmaf(w2.x, xv.x, sum);
        sum = fmaf(w2.y, xv.y, sum);
    }
    __shared__ float red[128];
    red[tid] = sum;
    __syncthreads();
    for (int s = 64; s > 0; s >>= 1) {
        if (tid < s) red[tid] += red[tid + s];
        __syncthreads();
    }
    if (tid == 0) y[row] = (red[0] + b[row]) * scale;
}

// ---------------------------------------------------------------------------
// Flash-decoding partial attention.
// grid = (NUM_HEADS, NCHUNK), block = 256 (8 waves). Each wave owns 64 keys
// (4 tiles of 16). Scores per tile via v_wmma_f32_16x16x4_f32 with A = q
// replicated over all 16 rows -> every lane of D holds score[lane&15].
// Online softmax per wave; partial (m, s, acc[64]) written to workspace.
// ---------------------------------------------------------------------------
__global__ void attn_partial_kernel(const float* __restrict__ Kc,
                                    const float* __restrict__ Vc,
                                    const float* __restrict__ q,     // [20*64], scaled
                                    float* __restrict__ m_buf,       // [20*PART]
                                    float* __restrict__ s_buf,       // [20*PART]
                                    float* __restrict__ acc_buf) {   // [20*PART*64]
    const int h      = blockIdx.x;
    const int lane   = threadIdx.x & 31;
    const int waveId = threadIdx.x >> 5;
    const int p      = blockIdx.y * WAVES_PER_BLOCK + waveId;   // partial index
    const int keyStart = p * KEYS_PER_WAVE;

    const int n     = lane & 15;          // key column within 16-key tile
    const int half2 = (lane >> 4) * 2;    // 0 or 2: which k-pair this lane holds

    // --- Preload A operand: A[m,k] = q[h, d0+k] replicated over m.
    // 32-bit operand lane layout: lanes 0-15 hold k={0,1} in (V0,V1),
    // lanes 16-31 hold k={2,3}. Same values for every lane in a half.
    const float* __restrict__ qh = q + h * HEAD_DIM;
    v2f aReg[16];
#pragma unroll
    for (int t = 0; t < 16; ++t) {
        const float2 qv = *reinterpret_cast<const float2*>(qh + t * 4 + half2);
        aReg[t].x = qv.x;
        aReg[t].y = qv.y;
    }

    const float* __restrict__ kHead = Kc + ((size_t)h * ENC_LEN) * HEAD_DIM;
    const float* __restrict__ vHead = Vc + ((size_t)h * ENC_LEN) * HEAD_DIM;

    float m = -INFINITY;
    float s = 0.f;
    float accX = 0.f, accY = 0.f;         // out[d] for d = 2*lane, 2*lane+1

#pragma unroll
    for (int tile = 0; tile < TILES_PER_WAVE; ++tile) {
        const int keyBase = keyStart + tile * 16;
        const float* __restrict__ kTile = kHead + (size_t)keyBase * HEAD_DIM;

        // --- B operand loads: B[k,n] = K[keyBase+n, d0+k]. Lane holds its
        // key's (d0+half2, d0+half2+1) pair as (V0,V1).
        v2f bReg[16];
#pragma unroll
        for (int t = 0; t < 16; ++t) {
            const float2 kv = *reinterpret_cast<const float2*>(
                kTile + (size_t)n * HEAD_DIM + t * 4 + half2);
            bReg[t].x = kv.x;
            bReg[t].y = kv.y;
        }

        // --- 16 chained WMMAs accumulate the 64-deep dot products.
        v8f acc = {};
#pragma unroll
        for (int t = 0; t < 16; ++t) {
            acc = __builtin_amdgcn_wmma_f32_16x16x4_f32(
                false, aReg[t], false, bReg[t], (short)0, acc, false, false);
        }
        const float score = acc[0];       // D[m,n] identical over m: lane -> key (lane&15)

        // --- Online softmax update (scores duplicated across halves; a
        // 4-step xor reduce within each 16-lane half gives the full result).
        float tmax = score;
#pragma unroll
        for (int off = 8; off >= 1; off >>= 1)
            tmax = fmaxf(tmax, __shfl_xor(tmax, off, 32));
        const float m_new = fmaxf(m, tmax);
        const float r = __expf(m - m_new);
        const float w = __expf(score - m_new);
        float wsum = w;
#pragma unroll
        for (int off = 8; off >= 1; off >>= 1)
            wsum += __shfl_xor(wsum, off, 32);
        s = s * r + wsum;
        accX *= r;
        accY *= r;

        // --- AV accumulation: per key a fully coalesced 256B float2 load.
        const float* __restrict__ vTile = vHead + (size_t)keyBase * HEAD_DIM;
#pragma unroll
        for (int i = 0; i < 16; ++i) {
            const float wi = __shfl(w, i, 32);
            const float2 vv = *reinterpret_cast<const float2*>(
                vTile + (size_t)i * HEAD_DIM + lane * 2);
            accX = fmaf(wi, vv.x, accX);
            accY = fmaf(wi, vv.y, accY);
        }
        m = m_new;
    }

    // --- Emit partial.
    const size_t pidx = (size_t)h * PART + p;
    if (lane == 0) {
        m_buf[pidx] = m;
        s_buf[pidx] = s;
    }
    float* __restrict__ ap = acc_buf + pidx * HEAD_DIM;
    ap[lane * 2 + 0] = accX;
    ap[lane * 2 + 1] = accY;
}

// ---------------------------------------------------------------------------
// Combine 128 partials per head: global max, exp-rescale, normalize.
// grid = 20 blocks, block = 128 threads.
// ---------------------------------------------------------------------------
__global__ void attn_combine_kernel(const float* __restrict__ m_buf,
                                    const float* __restrict__ s_buf,
                                    const float* __restrict__ acc_buf,
                                    float* __restrict__ attn_out) {  // [20*64]
    const int h = blockIdx.x;
    const int tid = threadIdx.x;

    __shared__ float sm[PART];
    __shared__ float se[PART];
    __shared__ float sred[PART];

    const float mv = m_buf[(size_t)h * PART + tid];
    sm[tid] = mv;
    __syncthreads();
    for (int s = 64; s > 0; s >>= 1) {
        if (tid < s) sm[tid] = fmaxf(sm[tid], sm[tid + s]);
        __syncthreads();
    }
    const float M = sm[0];

    const float e = __expf(mv - M);
    se[tid] = e;
    sred[tid] = e * s_buf[(size_t)h * PART + tid];
    __syncthreads();
    for (int s = 64; s > 0; s >>= 1) {
        if (tid < s) sred[tid] += sred[tid + s];
        __syncthreads();
    }
    const float S = sred[0];

    if (tid < HEAD_DIM) {
        float sum = 0.f;
        const float* __restrict__ ab = acc_buf + (size_t)h * PART * HEAD_DIM + tid;
#pragma unroll 8
        for (int p = 0; p < PART; ++p)
            sum = fmaf(se[p], ab[(size_t)p * HEAD_DIM], sum);
        attn_out[h * HEAD_DIM + tid] = sum / S;
    }
}

// ---------------------------------------------------------------------------
extern "C" void kernel_launch(void* const* d_in, const int* in_sizes, int n_in,
                              void* d_out, int out_size, void* d_ws, size_t ws_size,
                              hipStream_t stream) {
    const float* hs    = (const float*)d_in[0];   // [1,1,1280]
    const float* kc    = (const float*)d_in[1];   // [1,20,8192,64]
    const float* vc    = (const float*)d_in[2];   // [1,20,8192,64]
    const float* q_w   = (const float*)d_in[3];   // [1280,1280]
    const float* q_b   = (const float*)d_in[4];   // [1280]
    const float* out_w = (const float*)d_in[5];   // [1280,1280]
    const float* out_b = (const float*)d_in[6];   // [1280]
    float* out = (float*)d_out;                   // [1280]

    // Workspace layout (floats): ~686 KB total.
    float* q        = (float*)d_ws;                         // 1280
    float* m_buf    = q + EMBED_DIM;                        // 20*128
    float* s_buf    = m_buf + NUM_HEADS * PART;             // 20*128
    float* acc_buf  = s_buf + NUM_HEADS * PART;             // 20*128*64
    float* attn_out = acc_buf + (size_t)NUM_HEADS * PART * HEAD_DIM; // 1280

    // 1) q projection + scale
    gemv1280_kernel<<<EMBED_DIM, 128, 0, stream>>>(hs, q_w, q_b, q, ATTN_SCALE);

    // 2) flash-decoding partials (WMMA scores + online softmax)
    attn_partial_kernel<<<dim3(NUM_HEADS, NCHUNK), WAVES_PER_BLOCK * 32, 0, stream>>>(
        kc, vc, q, m_buf, s_buf, acc_buf);

    // 3) combine partials per head
    attn_combine_kernel<<<NUM_HEADS, PART, 0, stream>>>(m_buf, s_buf, acc_buf, attn_out);

    // 4) output projection + bias
    gemv1280_kernel<<<EMBED_DIM, 128, 0, stream>>>(attn_out, out_w, out_b, out, 1.0f);
}